// NoisyLabelLoss_25847113187334
// MI455X (gfx1250) — compile-verified
//
#include <hip/hip_runtime.h>
#include <hip/hip_bf16.h>
#include <stdint.h>

// Problem constants (fixed by the reference setup_inputs).
#define B_DIM    4
#define C_DIM    12
#define P_PIX    (512 * 512)            // pixels per image
#define N_PIX    (B_DIM * P_PIX)        // total pixels = 1,048,576
#define PQ       (P_PIX / 4)            // float4 quads per image
#define NTHREADS 256
#define NBLOCKS  (N_PIX / 4 / NTHREADS) // 1024 blocks, one quad per thread
#define IGN_IDX  255
#define ALPHA_C  0.1
#define EPS_C    1e-8f

// v_rcp_f32 + one Newton-Raphson step: ~3 VALU vs ~10 for strict fdiv,
// error ~ulp^2 (negligible for the three scalar outputs).
__device__ __forceinline__ float fast_rcp(float x) {
    float r = __builtin_amdgcn_rcpf(x);
    r = r * (2.0f - x * r);
    return r;
}

// Stage 1: streaming kernel. Each thread owns 4 consecutive pixels (float4
// loads -> global_load_b128, wave reads 512B contiguous per channel).
// Only the label row + column sums + diagonal of the per-pixel 12x12 CM are
// needed, so compute/registers stay tiny and the kernel is a pure HBM stream.
__global__ __launch_bounds__(NTHREADS)
void nll_main_kernel(const float* __restrict__ pred,
                     const float* __restrict__ cms,
                     const int*   __restrict__ labels,
                     float*       __restrict__ partials)
{
    const int tid = threadIdx.x;
    const int t   = blockIdx.x * NTHREADS + tid;   // pixel-quad id in [0, N_PIX/4)
    const int b   = t / PQ;
    const int p4  = t - b * PQ;

    // ---- gfx1250 async tensor path: stage this thread's 4 labels via
    //      GLOBAL_LOAD_ASYNC_TO_LDS_B128, tracked with ASYNCcnt. ----
    __shared__ int s_lbl[NTHREADS * 4];
    {
        unsigned lds_addr = (unsigned)(uintptr_t)(&s_lbl[tid * 4]);
        unsigned long long gaddr =
            (unsigned long long)(uintptr_t)(labels + (size_t)t * 4);
        asm volatile("global_load_async_to_lds_b128 %0, %1, off"
                     :: "v"(lds_addr), "v"(gaddr) : "memory");
        asm volatile("s_wait_asynccnt 0x0" ::: "memory");
    }
    const int l0 = s_lbl[tid * 4 + 0];
    const int l1 = s_lbl[tid * 4 + 1];
    const int l2 = s_lbl[tid * 4 + 2];
    const int l3 = s_lbl[tid * 4 + 3];

    const float4* __restrict__ cms4  = (const float4*)cms;
    const float4* __restrict__ pred4 = (const float4*)pred;
    const size_t cbase = (size_t)b * (C_DIM * C_DIM) * PQ + (size_t)p4;
    const size_t pbase = (size_t)b * C_DIM * PQ + (size_t)p4;

    float pn0 = 0.f, pn1 = 0.f, pn2 = 0.f, pn3 = 0.f;   // pred_noisy[label]
    float dg0 = 0.f, dg1 = 0.f, dg2 = 0.f, dg3 = 0.f;   // trace of normalized cm

    #pragma unroll
    for (int j = 0; j < C_DIM; ++j) {
        // gfx1250 global_prefetch_b8: pull next column's stream ahead.
        if (j + 1 < C_DIM)
            __builtin_prefetch((const void*)&cms4[cbase + (size_t)(j + 1) * PQ], 0, 3);

        float csx = 0.f, csy = 0.f, csz = 0.f, csw = 0.f;  // colsum_j
        float rvx = 0.f, rvy = 0.f, rvz = 0.f, rvw = 0.f;  // cm[lbl, j] (raw)
        float dx = 0.f, dy = 0.f, dz = 0.f, dw = 0.f;      // cm[j, j] (raw)
        #pragma unroll
        for (int i = 0; i < C_DIM; ++i) {
            float4 v = cms4[cbase + (size_t)(i * C_DIM + j) * PQ];
            csx += v.x; csy += v.y; csz += v.z; csw += v.w;
            if (i == j) { dx = v.x; dy = v.y; dz = v.z; dw = v.w; } // compile-time
            rvx = (l0 == i) ? v.x : rvx;
            rvy = (l1 == i) ? v.y : rvy;
            rvz = (l2 == i) ? v.z : rvz;
            rvw = (l3 == i) ? v.w : rvw;
        }
        float4 pj = pred4[pbase + (size_t)j * PQ];
        const float ix = fast_rcp(csx), iy = fast_rcp(csy),
                    iz = fast_rcp(csz), iw = fast_rcp(csw);
        pn0 += rvx * pj.x * ix;  pn1 += rvy * pj.y * iy;
        pn2 += rvz * pj.z * iz;  pn3 += rvw * pj.w * iw;
        dg0 += dx * ix;  dg1 += dy * iy;  dg2 += dz * iz;  dg3 += dw * iw;
    }

    float ce = 0.f;
    int   vc = 0;
    if (l0 != IGN_IDX) { ce -= logf(pn0 + EPS_C); ++vc; }
    if (l1 != IGN_IDX) { ce -= logf(pn1 + EPS_C); ++vc; }
    if (l2 != IGN_IDX) { ce -= logf(pn2 + EPS_C); ++vc; }
    if (l3 != IGN_IDX) { ce -= logf(pn3 + EPS_C); ++vc; }
    const float dsum = dg0 + dg1 + dg2 + dg3;

    // ---- deterministic block tree reduction in LDS ----
    __shared__ float r_ce[NTHREADS];
    __shared__ float r_dg[NTHREADS];
    __shared__ float r_vc[NTHREADS];
    r_ce[tid] = ce;
    r_dg[tid] = dsum;
    r_vc[tid] = (float)vc;
    __syncthreads();
    #pragma unroll
    for (int s = NTHREADS / 2; s > 0; s >>= 1) {
        if (tid < s) {
            r_ce[tid] += r_ce[tid + s];
            r_dg[tid] += r_dg[tid + s];
            r_vc[tid] += r_vc[tid + s];
        }
        __syncthreads();
    }
    if (tid == 0) {
        partials[blockIdx.x * 3 + 0] = r_ce[0];
        partials[blockIdx.x * 3 + 1] = r_dg[0];
        partials[blockIdx.x * 3 + 2] = r_vc[0];
    }
}

// Stage 2: single-block deterministic finalize (double accumulation, fixed
// order). Writes (loss, loss_ce, regularisation) as 3 floats.
__global__ __launch_bounds__(256)
void nll_finalize_kernel(const float* __restrict__ partials,
                         float* __restrict__ out)
{
    const int tid = threadIdx.x;
    __shared__ double s_ce[256];
    __shared__ double s_dg[256];
    __shared__ double s_vc[256];
    double ce = 0.0, dg = 0.0, vc = 0.0;
    for (int i = tid; i < NBLOCKS; i += 256) {
        ce += (double)partials[i * 3 + 0];
        dg += (double)partials[i * 3 + 1];
        vc += (double)partials[i * 3 + 2];
    }
    s_ce[tid] = ce; s_dg[tid] = dg; s_vc[tid] = vc;
    __syncthreads();
    #pragma unroll
    for (int s = 128; s > 0; s >>= 1) {
        if (tid < s) {
            s_ce[tid] += s_ce[tid + s];
            s_dg[tid] += s_dg[tid + s];
            s_vc[tid] += s_vc[tid + s];
        }
        __syncthreads();
    }
    if (tid == 0) {
        const double nvalid  = (s_vc[0] < 1.0) ? 1.0 : s_vc[0];
        const double loss_ce = s_ce[0] / nvalid;
        const double reg     = ALPHA_C * (s_dg[0] / ((double)B_DIM * (double)P_PIX));
        const double loss    = loss_ce - reg;   // MIN_TRACE == False
        out[0] = (float)loss;
        out[1] = (float)loss_ce;
        out[2] = (float)reg;
    }
}

extern "C" void kernel_launch(void* const* d_in, const int* in_sizes, int n_in,
                              void* d_out, int out_size, void* d_ws, size_t ws_size,
                              hipStream_t stream) {
    const float* pred   = (const float*)d_in[0];
    const float* cms    = (const float*)d_in[1];
    const int*   labels = (const int*)d_in[2];   // integer inputs arrive as int32
    float* partials = (float*)d_ws;              // NBLOCKS * 3 floats = 12 KB
    float* out      = (float*)d_out;             // 3 floats

    nll_main_kernel<<<NBLOCKS, NTHREADS, 0, stream>>>(pred, cms, labels, partials);
    nll_finalize_kernel<<<1, 256, 0, stream>>>(partials, out);
    (void)in_sizes; (void)n_in; (void)out_size; (void)ws_size;
}